// CMLayer_61194694034213
// MI455X (gfx1250) — compile-verified
//
#include <hip/hip_runtime.h>
#include <math.h>
#include <stdint.h>

// CMLayer forward, restructured:
//   [A^-1]_ports = (-j*I2 - T(w))^-1,  T built from one real-symmetric 8x8
//   eigendecomposition per batch element (parallel-order Jacobi), then O(8)
//   work per (b,f). Kernel 2 is output-bandwidth bound
//   (13.1 MB @ 23.3 TB/s ~ 0.6 us floor); kernel 1 is latency-bound, so it
//   uses 4-way ILP round-robin rotations + single-instruction trans ops.

namespace {
constexpr int kB = 1024;       // batch
constexpr int kL = 19;         // params per batch element
constexpr int kF = 401;        // frequencies
constexpr int kEigStride = 32; // floats per batch record in workspace (128 B)
constexpr int kSweeps = 8;     // parallel-order Jacobi sweeps (8x8)
}

// ------------------------------------------------------------------
// Kernel 2 (listed first so the disasm snippet shows its prologue):
// per-(b,f) S-matrix. One wave async-stages the 32-float batch record
// into LDS (CDNA5 ASYNCcnt path), then all threads do O(8) flops and
// two coalesced global_store_b128.
// ------------------------------------------------------------------
__global__ __launch_bounds__(256) void cm_freq_sweep(const float* __restrict__ eig,
                                                     const float* __restrict__ om,
                                                     float* __restrict__ out) {
  __shared__ float sd[kEigStride];
  const int b = blockIdx.x;

  if (threadIdx.x < kEigStride) {  // exactly one wave32
    const float* g = eig + (size_t)b * kEigStride + threadIdx.x;
#if defined(__gfx1250__) && __has_builtin(__builtin_amdgcn_global_load_async_to_lds_b32)
    typedef __attribute__((address_space(1))) int gint_t;
    typedef __attribute__((address_space(3))) int lint_t;
    __builtin_amdgcn_global_load_async_to_lds_b32(
        (gint_t*)(uintptr_t)g,
        (lint_t*)(uint32_t)(uintptr_t)&sd[threadIdx.x], 0, 0);
#if __has_builtin(__builtin_amdgcn_s_wait_asynccnt)
    __builtin_amdgcn_s_wait_asynccnt(0);
#else
    asm volatile("s_wait_asynccnt 0" ::: "memory");
#endif
#else
    sd[threadIdx.x] = *g;
#endif
  }
  __syncthreads();

  const int f = blockIdx.y * blockDim.x + threadIdx.x;
  if (f >= kF) return;
  const float w = om[f];

  float T11 = 0.0f, T22 = 0.0f, T12 = 0.0f;
#pragma unroll
  for (int k = 0; k < 8; ++k) {
    const float inv = 1.0f / (w + sd[k]);  // exact div: kernel 2 has slack
    const float a = sd[8 + k];
    const float bb = sd[16 + k];
    T11 = fmaf(a * a, inv, T11);
    T22 = fmaf(bb * bb, inv, T22);
    T12 = fmaf(a * bb, inv, T12);
  }

  // Y = (-j*I - T)^-1 ; S = I + 2j*Y ; det = (T11*T22 - T12^2 - 1) + j(T11+T22)
  const float dr = fmaf(T11, T22, -fmaf(T12, T12, 1.0f));
  const float di = T11 + T22;
  const float idet = 1.0f / fmaf(dr, dr, di * di);

  const float s11r = 1.0f + (2.0f * dr - 2.0f * T22 * di) * idet;
  const float s11i = (-2.0f * T22 * dr - 2.0f * di) * idet;
  const float s22r = 1.0f + (2.0f * dr - 2.0f * T11 * di) * idet;
  const float s22i = (-2.0f * T11 * dr - 2.0f * di) * idet;
  const float s12r = 2.0f * T12 * di * idet;
  const float s12i = 2.0f * T12 * dr * idet;

  // Output: (B, F, 2, 2) complex64, interleaved re/im -> 8 floats = 32 B/thread.
  float4* o4 = (float4*)(out + (((size_t)b * kF + f) * 8));
  o4[0] = make_float4(s11r, s11i, s12r, s12i);
  o4[1] = make_float4(s12r, s12i, s22r, s22i);  // S21 == S12 (T symmetric)
}

// ------------------------------------------------------------------
// Kernel 1: per-batch 8x8 real symmetric Jacobi eigendecomposition.
// Parallel (round-robin) ordering: each sweep = 7 rounds of 4 disjoint
// rotations; the 4 c/s chains per round are independent (ILP hides
// v_rcp/v_rsq/v_sqrt latency). Only rows 0 and 7 of Q are accumulated.
// Record layout per batch b (stride 32 floats):
//   [0..7]   lambda_k
//   [8..15]  a_k = m01 * Q[0][k]
//   [16..23] b_k = m89 * Q[7][k]
// ------------------------------------------------------------------
__global__ __launch_bounds__(32) void cm_eig_jacobi(const float* __restrict__ pv,
                                                    float* __restrict__ eig) {
  const int b = blockIdx.x * blockDim.x + threadIdx.x;
  if (b >= kB) return;

  float p[kL];
#pragma unroll
  for (int i = 0; i < kL; ++i) p[i] = pv[b * kL + i];

  // Inner coupling matrix M_II (nodes 1..8 -> indices 0..7).
  // Params: 0:(0,1)  1..7:(1,2)..(7,8)  8:(8,9)  9..16:diag(1,1)..(8,8)
  //         17:(2,7)->inner(1,6)  18:(3,6)->inner(2,5)
  float A[8][8];
#pragma unroll
  for (int i = 0; i < 8; ++i)
#pragma unroll
    for (int j = 0; j < 8; ++j) A[i][j] = 0.0f;
#pragma unroll
  for (int i = 0; i < 8; ++i) A[i][i] = p[9 + i];
#pragma unroll
  for (int i = 0; i < 7; ++i) { A[i][i + 1] = p[1 + i]; A[i + 1][i] = p[1 + i]; }
  A[1][6] = p[17]; A[6][1] = p[17];
  A[2][5] = p[18]; A[5][2] = p[18];

  float Q0[8], Q7[8];  // rows 0 and 7 of the accumulated eigenvector matrix
#pragma unroll
  for (int k = 0; k < 8; ++k) { Q0[k] = (k == 0) ? 1.0f : 0.0f; Q7[k] = (k == 7) ? 1.0f : 0.0f; }

#pragma unroll 1
  for (int sweep = 0; sweep < kSweeps; ++sweep) {
#pragma unroll
    for (int r = 0; r < 7; ++r) {  // round-robin tournament rounds
      // Disjoint pairs this round (circle method, player 0 fixed):
      //   g0: (0, x0)  g1: (x1, x6)  g2: (x2, x5)  g3: (x3, x4),
      //   x_i = 1 + (i + r) % 7. All indices fold to constants.
      int PP[4], QQ[4];
      PP[0] = 0;                 QQ[0] = 1 + (0 + r) % 7;
      PP[1] = 1 + (1 + r) % 7;   QQ[1] = 1 + (6 + r) % 7;
      PP[2] = 1 + (2 + r) % 7;   QQ[2] = 1 + (5 + r) % 7;
      PP[3] = 1 + (3 + r) % 7;   QQ[3] = 1 + (4 + r) % 7;

      // Phase 1: four independent c/s/t chains (trans-op latency overlapped).
      float cg[4], sg[4], tg[4], apqg[4];
#pragma unroll
      for (int g = 0; g < 4; ++g) {
        const int pp = PP[g], qq = QQ[g];
        const float apq = A[pp][qq];
        apqg[g] = apq;
        float c = 1.0f, s = 0.0f, t = 0.0f;
        if (fabsf(apq) > 1e-12f) {
          const float tau = (A[qq][qq] - A[pp][pp]) *
                            (0.5f * __builtin_amdgcn_rcpf(apq));
          t = copysignf(1.0f, tau) *
              __builtin_amdgcn_rcpf(fabsf(tau) +
                                    __builtin_amdgcn_sqrtf(fmaf(tau, tau, 1.0f)));
          c = __builtin_amdgcn_rsqf(fmaf(t, t, 1.0f));
          s = t * c;
        }
        cg[g] = c; sg[g] = s; tg[g] = t;
      }

      // Phase 2: apply the four (commuting) rotations. Symmetric update:
      // only rows/cols p,q touched; diagonal via a'pp = app - t*apq.
#pragma unroll
      for (int g = 0; g < 4; ++g) {
        const int pp = PP[g], qq = QQ[g];
        const float c = cg[g], s = sg[g], t = tg[g], apq = apqg[g];
#pragma unroll
        for (int k = 0; k < 8; ++k) {
          if (k == pp || k == qq) continue;
          const float akp = A[k][pp], akq = A[k][qq];
          const float nkp = c * akp - s * akq;
          const float nkq = s * akp + c * akq;
          A[k][pp] = nkp; A[pp][k] = nkp;
          A[k][qq] = nkq; A[qq][k] = nkq;
        }
        A[pp][pp] = fmaf(-t, apq, A[pp][pp]);
        A[qq][qq] = fmaf(t, apq, A[qq][qq]);
        A[pp][qq] = 0.0f; A[qq][pp] = 0.0f;
        const float q0p = Q0[pp], q0q = Q0[qq];
        Q0[pp] = c * q0p - s * q0q; Q0[qq] = s * q0p + c * q0q;
        const float q7p = Q7[pp], q7q = Q7[qq];
        Q7[pp] = c * q7p - s * q7q; Q7[qq] = s * q7p + c * q7q;
      }
    }
  }

  float* o = eig + (size_t)b * kEigStride;
#pragma unroll
  for (int k = 0; k < 8; ++k) o[k] = A[k][k];
#pragma unroll
  for (int k = 0; k < 8; ++k) o[8 + k] = p[0] * Q0[k];
#pragma unroll
  for (int k = 0; k < 8; ++k) o[16 + k] = p[8] * Q7[k];
#pragma unroll
  for (int k = 24; k < 32; ++k) o[k] = 0.0f;
}

// ------------------------------------------------------------------
extern "C" void kernel_launch(void* const* d_in, const int* in_sizes, int n_in,
                              void* d_out, int out_size, void* d_ws, size_t ws_size,
                              hipStream_t stream) {
  (void)in_sizes; (void)n_in; (void)out_size; (void)ws_size;
  const float* pv = (const float*)d_in[0];  // param_vec (B, 19) f32
  const float* om = (const float*)d_in[1];  // omega (401,) f32
  // d_in[2]/d_in[3] (rows/cols) are module-level constants in the reference;
  // their scatter pattern is baked into cm_eig_jacobi.
  float* out = (float*)d_out;
  float* eig = (float*)d_ws;  // kB * kEigStride floats = 128 KB scratch

  // Latency-bound eigensolve: 1 wave per workgroup, spread across WGPs.
  cm_eig_jacobi<<<dim3(kB / 32), dim3(32), 0, stream>>>(pv, eig);
  // Bandwidth-bound sweep: 2048 blocks x 256 threads.
  cm_freq_sweep<<<dim3(kB, (kF + 255) / 256), dim3(256), 0, stream>>>(eig, om, out);
}